// DCRNNEdgePredictor_15255723835743
// MI455X (gfx1250) — compile-verified
//
#include <hip/hip_runtime.h>
#include <math.h>

#define N_NODES 1024
#define N_EDGES 32768
#define PAIRS   (N_NODES * (N_NODES - 1))   // 1,047,552 = 4092 * 256

// workspace layout (in floats)
#define WS_DEGOUT 0
#define WS_DEGIN  1024
#define WS_TXO    2048          // 1024 x 8
#define WS_TXI    10240         // 1024 x 8
#define WS_OUTCAT 18432         // 1024 x 40
#define WS_APAD   59392         // 1024 x 8 (cols 0..4 used)
#define WS_BPAD   67584         // 1024 x 8 (cols 0..4 used)
#define WS_PACK   75776         // 80 packed scalars

typedef float v2f __attribute__((ext_vector_type(2)));
typedef float v8f __attribute__((ext_vector_type(8)));

__device__ __forceinline__ v8f wmma_f32_4(v2f a, v2f b, v8f c) {
  // V_WMMA_F32_16X16X4_F32 : exact f32 MACC, 16x16 tile
  return __builtin_amdgcn_wmma_f32_16x16x4_f32(false, a, false, b, (short)0, c, false, false);
}

// ---------------- init / pack ----------------
__global__ void k_init(float* ws) {
  int t = blockIdx.x * blockDim.x + threadIdx.x;   // 72*256 = 18432 covers deg+Txo+Txi
  ws[t] = 0.0f;
}

__global__ void k_pack(const float* __restrict__ w1, const float* __restrict__ b1,
                       const float* __restrict__ w2, const float* __restrict__ b2,
                       const float* __restrict__ w3, const float* __restrict__ b3,
                       const float* __restrict__ w4, const float* __restrict__ b4,
                       float* __restrict__ pp) {
  int t = threadIdx.x;
  if (t >= 80) return;
  float v = 0.0f;
  if (t < 45)       v = w1[80 * 5 + t];      // w1 rows 80..88 (dist,d2,lags,lz,atd,ato,dta,ota,vr)
  else if (t < 50)  v = b1[t - 45];
  else if (t < 65)  v = w2[t - 50];          // 5x3
  else if (t < 68)  v = b2[t - 65];
  else if (t < 71)  v = w3[t - 68];          // 3x1
  else if (t == 71) v = b3[0];
  else if (t < 74)  v = w4[t - 72];          // 2x1
  else if (t == 74) v = b4[0];
  pp[t] = v;
}

// ---------------- graph stages ----------------
__global__ void k_deg(const int* __restrict__ ei, const float* __restrict__ ew,
                      float* __restrict__ deg_out, float* __restrict__ deg_in) {
  int e = blockIdx.x * blockDim.x + threadIdx.x;
  int r = ei[e], c = ei[N_EDGES + e];
  float w = ew[e];
  atomicAdd(&deg_out[r], w);
  atomicAdd(&deg_in[c], w);
}

__global__ void k_prop(const int* __restrict__ ei, const float* __restrict__ x,
                       const float* __restrict__ deg_out, const float* __restrict__ deg_in,
                       float* __restrict__ Txo, float* __restrict__ Txi) {
  int e = blockIdx.x * blockDim.x + threadIdx.x;
  int r = ei[e], c = ei[N_EDGES + e];
  float no = 1.0f / deg_out[r];
  float ni = 1.0f / deg_in[r];      // reference: norm_in = (1/deg_in)[row]
#pragma unroll
  for (int k = 0; k < 8; ++k) {
    atomicAdd(&Txo[c * 8 + k], no * x[r * 8 + k]);  // pf: out[col] += no * x[row]
    atomicAdd(&Txi[r * 8 + k], ni * x[c * 8 + k]);  // pb: out[row] += ni * x[col]
  }
}

// ---------------- WMMA GRU ----------------
__device__ __forceinline__ float wcomb(const float* __restrict__ W, int k, int n) {
  // W: [2][2][40][32]; combined 24x32: rows0-7 = W[0,0]+W[1,0], 8-15 = W[0,1], 16-23 = W[1,1]
  if (k < 8)       return W[k * 32 + n] + W[2 * 1280 + k * 32 + n];
  else if (k < 16) return W[1280 + (k - 8) * 32 + n];
  else             return W[3 * 1280 + (k - 16) * 32 + n];
}

__device__ __forceinline__ float loadF(const float* __restrict__ x, const float* __restrict__ Txo,
                                       const float* __restrict__ Txi, int row, int k) {
  if (k < 8)       return x[row * 8 + k];
  else if (k < 16) return Txo[row * 8 + (k - 8)];
  else             return Txi[row * 8 + (k - 16)];
}

__global__ void k_gru(const float* __restrict__ x,
                      const float* __restrict__ Wz, const float* __restrict__ bz,
                      const float* __restrict__ Wh, const float* __restrict__ bh,
                      const float* __restrict__ Txo, const float* __restrict__ Txi,
                      float* __restrict__ outcat) {
  int wave = (blockIdx.x * blockDim.x + threadIdx.x) >> 5;  // 64 waves, one 16-row tile each
  int lane = threadIdx.x & 31;
  int m0 = wave * 16;
  int half = lane >> 4, lm = lane & 15;
  int mrow = m0 + lm;
  int koff = half * 2;

  v8f cz0 = {}, cz1 = {}, ch0 = {}, ch1 = {};
#pragma unroll
  for (int ks = 0; ks < 6; ++ks) {           // K = 24, steps of 4
    int ka = ks * 4 + koff;
    v2f a;  a.x = loadF(x, Txo, Txi, mrow, ka); a.y = loadF(x, Txo, Txi, mrow, ka + 1);
    v2f bz0v, bz1v, bh0v, bh1v;
    bz0v.x = wcomb(Wz, ka, lm);       bz0v.y = wcomb(Wz, ka + 1, lm);
    bz1v.x = wcomb(Wz, ka, 16 + lm);  bz1v.y = wcomb(Wz, ka + 1, 16 + lm);
    bh0v.x = wcomb(Wh, ka, lm);       bh0v.y = wcomb(Wh, ka + 1, lm);
    bh1v.x = wcomb(Wh, ka, 16 + lm);  bh1v.y = wcomb(Wh, ka + 1, 16 + lm);
    cz0 = wmma_f32_4(a, bz0v, cz0);
    cz1 = wmma_f32_4(a, bz1v, cz1);
    ch0 = wmma_f32_4(a, bh0v, ch0);
    ch1 = wmma_f32_4(a, bh1v, ch1);
  }
  // epilogue: H = (1 - sigmoid(Hz)) * tanh(Hh); out_cat = [relu(H) | x]
#pragma unroll
  for (int v = 0; v < 8; ++v) {
    int row = m0 + half * 8 + v;
    float hz0 = cz0[v] + bz[lm];
    float hz1 = cz1[v] + bz[16 + lm];
    float th0 = tanhf(ch0[v] + bh[lm]);
    float th1 = tanhf(ch1[v] + bh[16 + lm]);
    float z0 = 1.0f / (1.0f + expf(-hz0));
    float z1 = 1.0f / (1.0f + expf(-hz1));
    outcat[row * 40 + lm]      = fmaxf((1.0f - z0) * th0, 0.0f);
    outcat[row * 40 + 16 + lm] = fmaxf((1.0f - z1) * th1, 0.0f);
  }
#pragma unroll
  for (int c = 0; c < 4; ++c) {
    int cc = half * 4 + c;
    outcat[mrow * 40 + 32 + cc] = x[mrow * 8 + cc];
  }
}

// ---------------- WMMA projection: out_cat(1024x40) @ [w1[0:40] | w1[40:80]] -> A5,B5 ----------------
__device__ __forceinline__ float wab(const float* __restrict__ w1, int k, int n) {
  if (n < 5)       return w1[k * 5 + n];
  else if (n < 10) return w1[(40 + k) * 5 + (n - 5)];
  else             return 0.0f;
}

__global__ void k_proj(const float* __restrict__ outcat, const float* __restrict__ w1,
                       float* __restrict__ Apad, float* __restrict__ Bpad) {
  int wave = (blockIdx.x * blockDim.x + threadIdx.x) >> 5;
  int lane = threadIdx.x & 31;
  int m0 = wave * 16;
  int half = lane >> 4, lm = lane & 15;
  int mrow = m0 + lm;
  int koff = half * 2;

  v8f c = {};
#pragma unroll
  for (int ks = 0; ks < 10; ++ks) {          // K = 40
    int ka = ks * 4 + koff;
    v2f a;  a.x = outcat[mrow * 40 + ka]; a.y = outcat[mrow * 40 + ka + 1];
    v2f b;  b.x = wab(w1, ka, lm);        b.y = wab(w1, ka + 1, lm);
    c = wmma_f32_4(a, b, c);
  }
#pragma unroll
  for (int v = 0; v < 8; ++v) {
    int row = m0 + half * 8 + v;
    if (lm < 5)       Apad[row * 8 + lm] = c[v];
    else if (lm < 10) Bpad[row * 8 + (lm - 5)] = c[v];
  }
}

// ---------------- streaming pair MLP (memory-bound bulk) ----------------
__global__ void __launch_bounds__(256) k_pairs(
    const float* __restrict__ dist, const float* __restrict__ lags,
    const float* __restrict__ lag_zeros, const float* __restrict__ alltod,
    const float* __restrict__ alltoo, const float* __restrict__ dtoall,
    const float* __restrict__ otoall, const float* __restrict__ valreverse,
    const float* __restrict__ Apad, const float* __restrict__ Bpad,
    const float* __restrict__ pp, float* __restrict__ out) {
  int p = blockIdx.x * 256 + threadIdx.x;

  float P_[80];
  const float4* pv = (const float4*)pp;
#pragma unroll
  for (int t = 0; t < 20; ++t) {
    float4 v = pv[t];
    P_[4 * t] = v.x; P_[4 * t + 1] = v.y; P_[4 * t + 2] = v.z; P_[4 * t + 3] = v.w;
  }

  unsigned up = (unsigned)p;
  unsigned i = up / 1023u;
  unsigned r = up - i * 1023u;
  unsigned j = r + (r >= i ? 1u : 0u);

  float4 av = *(const float4*)(Apad + i * 8); float a4 = Apad[i * 8 + 4];
  float4 bv = *(const float4*)(Bpad + j * 8); float b4 = Bpad[j * 8 + 4];
  float A_[5] = {av.x, av.y, av.z, av.w, a4};
  float B_[5] = {bv.x, bv.y, bv.z, bv.w, b4};

  float d  = dist[p];
  float lg = lags[p], lz = lag_zeros[p];
  float feat[9] = {d, d * d, lg, lz, alltod[p], alltoo[p], dtoall[p], otoall[p], valreverse[p]};

  float h1[5];
#pragma unroll
  for (int c = 0; c < 5; ++c) {
    float s = A_[c] + B_[c] + P_[45 + c];
#pragma unroll
    for (int k = 0; k < 9; ++k) s += feat[k] * P_[k * 5 + c];
    h1[c] = fmaxf(s, 0.0f);
  }
  float h2[3];
#pragma unroll
  for (int c = 0; c < 3; ++c) {
    float s = P_[65 + c];
#pragma unroll
    for (int k = 0; k < 5; ++k) s += h1[k] * P_[50 + k * 3 + c];
    h2[c] = fmaxf(s, 0.0f);
  }
  float h3 = P_[71] + h2[0] * P_[68] + h2[1] * P_[69] + h2[2] * P_[70];
  out[p] = h3 * P_[72] + h3 * lz * P_[73] + P_[74];
}

// ---------------- launcher ----------------
extern "C" void kernel_launch(void* const* d_in, const int* in_sizes, int n_in,
                              void* d_out, int out_size, void* d_ws, size_t ws_size,
                              hipStream_t stream) {
  const float* x    = (const float*)d_in[0];
  const int*   ei   = (const int*)  d_in[1];
  const float* ew   = (const float*)d_in[2];
  const float* dist = (const float*)d_in[3];
  const float* lags = (const float*)d_in[4];
  const float* lagz = (const float*)d_in[5];
  const float* atd  = (const float*)d_in[6];
  const float* ato  = (const float*)d_in[7];
  const float* dta  = (const float*)d_in[8];
  const float* ota  = (const float*)d_in[9];
  const float* vr   = (const float*)d_in[10];
  const float* Wz = (const float*)d_in[11]; const float* bz = (const float*)d_in[12];
  // Wr (13), br (14) unused: R*H0 == 0 since H0 == 0
  const float* Wh = (const float*)d_in[15]; const float* bh = (const float*)d_in[16];
  const float* w1 = (const float*)d_in[17]; const float* b1 = (const float*)d_in[18];
  const float* w2 = (const float*)d_in[19]; const float* b2 = (const float*)d_in[20];
  const float* w3 = (const float*)d_in[21]; const float* b3 = (const float*)d_in[22];
  const float* w4 = (const float*)d_in[23]; const float* b4 = (const float*)d_in[24];

  float* ws     = (float*)d_ws;
  float* degO   = ws + WS_DEGOUT;
  float* degI   = ws + WS_DEGIN;
  float* Txo    = ws + WS_TXO;
  float* Txi    = ws + WS_TXI;
  float* outcat = ws + WS_OUTCAT;
  float* Apad   = ws + WS_APAD;
  float* Bpad   = ws + WS_BPAD;
  float* pp     = ws + WS_PACK;

  k_init <<<72, 256, 0, stream>>>(ws);                       // zero deg + Txo + Txi
  k_pack <<<1, 128, 0, stream>>>(w1, b1, w2, b2, w3, b3, w4, b4, pp);
  k_deg  <<<N_EDGES / 256, 256, 0, stream>>>(ei, ew, degO, degI);
  k_prop <<<N_EDGES / 256, 256, 0, stream>>>(ei, x, degO, degI, Txo, Txi);
  k_gru  <<<8, 256, 0, stream>>>(x, Wz, bz, Wh, bh, Txo, Txi, outcat);
  k_proj <<<8, 256, 0, stream>>>(outcat, w1, Apad, Bpad);
  k_pairs<<<PAIRS / 256, 256, 0, stream>>>(dist, lags, lagz, atd, ato, dta, ota, vr,
                                           Apad, Bpad, pp, (float*)d_out);
}